// GCN_71244917506308
// MI455X (gfx1250) — compile-verified
//
#include <hip/hip_runtime.h>

typedef __attribute__((ext_vector_type(2)))  float        v2f;
typedef __attribute__((ext_vector_type(8)))  float        v8f;
typedef __attribute__((ext_vector_type(4)))  unsigned int v4u;
typedef __attribute__((ext_vector_type(8)))  int          v8i;
typedef __attribute__((ext_vector_type(4)))  int          v4i;

// ---------------------------------------------------------------------------
// zero fill (float4 vectorized; all our buffers are multiples of 128 floats)
// ---------------------------------------------------------------------------
__global__ void gcn_zero_f32v4(float4* __restrict__ p, long long n4) {
  long long i = (long long)blockIdx.x * blockDim.x + threadIdx.x;
  if (i < n4) p[i] = make_float4(0.f, 0.f, 0.f, 0.f);
}

// ---------------------------------------------------------------------------
// SpMM scatter: out[rows[e], :] += vals[e] * h[cols[e], :]
// One wave32 per edge; lane l handles float4 #l of the row. Gathers are
// coalesced 512B row segments; scatters are global_atomic_add_f32 landing in
// the 192MB L2 (whole output matrix is L2-resident at 51-102 MB).
// ---------------------------------------------------------------------------
template<int D>
__global__ __launch_bounds__(256)
void gcn_spmm_scatter(const int*   __restrict__ rows,
                      const int*   __restrict__ cols,
                      const float* __restrict__ vals,
                      const float* __restrict__ h,
                      float*       __restrict__ out,
                      int nnz)
{
  const int e = blockIdx.x * 8 + (threadIdx.x >> 5);
  if (e >= nnz) return;
  const int lane = threadIdx.x & 31;
  const int r = rows[e];
  const int c = cols[e];
  const float v = vals[e];
  const float4* __restrict__ src = (const float4*)(h + (long long)c * D);
  float* __restrict__ dst = out + (long long)r * D;
#pragma unroll
  for (int i = 0; i < D / 128; ++i) {
    float4 t = src[lane + 32 * i];
    const int b = (lane + 32 * i) * 4;
    atomicAdd(dst + b + 0, v * t.x);
    atomicAdd(dst + b + 1, v * t.y);
    atomicAdd(dst + b + 2, v * t.z);
    atomicAdd(dst + b + 3, v * t.w);
  }
}

// ---------------------------------------------------------------------------
// TDM: issue a 2D tile load (tileW x tileH elements of 4B, row stride
// `strideElems`) from global into LDS at byte offset `ldsOff`, with hardware
// padding of 1 dword every 32 dwords -> LDS row stride = 33 floats
// (conflict-free for our fragment reads). D# per CDNA5 ISA ch.8.
// This toolchain exposes the 6-arg builtin:
//   (uint32x4 g0, int32x8 g1, int32x4 g2, int32x4 g3, int32x8 ext, i32 cpol)
// ---------------------------------------------------------------------------
#if __has_builtin(__builtin_amdgcn_tensor_load_to_lds)
#define GCN_HAVE_TDM 1
__device__ __forceinline__ void gcn_tdm_load_2d(const float* gptr,
                                                unsigned ldsOff,
                                                unsigned tileW, unsigned tileH,
                                                unsigned tensorH,
                                                unsigned long long strideElems)
{
  const unsigned long long ga = (unsigned long long)(uintptr_t)gptr;
  v4u g0;
  g0[0] = 1u;                                       // count=1 (valid user D#)
  g0[1] = ldsOff;                                   // lds_addr (bytes)
  g0[2] = (unsigned)(ga & 0xFFFFFFFFu);             // global_addr[31:0]
  g0[3] = (unsigned)(ga >> 32) | (2u << 30);        // global_addr[56:32] | type=2
  v8i g1;
  // data_size=2 (4B) | pad_enable | pad_interval=4 (32 dwords) | pad_amount=0 (1 dword)
  g1[0] = (int)((2u << 16) | (1u << 20) | (4u << 22));
  g1[1] = (int)((tileW & 0xFFFFu) << 16);                        // tensor_dim0[15:0]
  g1[2] = (int)((tileW >> 16) | ((tensorH & 0xFFFFu) << 16));    // td0[31:16] | td1[15:0]
  g1[3] = (int)((tensorH >> 16) | ((tileW & 0xFFFFu) << 16));    // td1[31:16] | tile_dim0
  g1[4] = (int)(tileH & 0xFFFFu);                                // tile_dim1 (tile_dim2=0)
  g1[5] = (int)(strideElems & 0xFFFFFFFFull);                    // dim0_stride[31:0]
  g1[6] = (int)((strideElems >> 32) & 0xFFFFull);                // dim0_stride[47:32]
  g1[7] = 0;
  const v4i z4 = {0, 0, 0, 0};
  const v8i z8 = {0, 0, 0, 0, 0, 0, 0, 0};
  __builtin_amdgcn_tensor_load_to_lds(g0, g1, z4, z4, z8, 0);
}
#endif

// ---------------------------------------------------------------------------
// Dense GEMM via WMMA:  O[M x NTOT] = act( H[M x K] * W[NTOT x K]^T )
// Block = 256 threads = 8 waves; block tile = 16 rows x 128 cols, one 16x16
// C tile per wave. A (16x32) and B (128 n-rows x 32 k) staged per K-chunk:
// TDM descriptor loads (wave 0 issues, TENSORcnt-waited) with HW-padded rows
// (stride 33 floats); fallback = cooperative vector loads. Compute uses
// V_WMMA_F32_16X16X4_F32 (full f32 precision), 8 per chunk.
// ---------------------------------------------------------------------------
template<int K, int NTOT, bool RELU>
__global__ __launch_bounds__(256)
void gcn_gemm_wmma(const float* __restrict__ H,
                   const float* __restrict__ W,   // [NTOT x K] row major
                   float*       __restrict__ O,
                   int M)
{
  constexpr int KC = 32;
  constexpr int RS = 33;                       // padded row stride (floats)
  __shared__ __attribute__((aligned(16))) float As[16 * RS];   // As[r][k]
  __shared__ __attribute__((aligned(16))) float Bs[128 * RS];  // Bs[n][k]

  const int tid   = threadIdx.x;
  const int wave  = tid >> 5;
  const int lane  = tid & 31;
  const int lh    = lane >> 4;                 // lane half
  const int mrow  = lane & 15;
  const int mBase = blockIdx.x * 16;
  const int nBase = blockIdx.y * 128;
  const int ncol  = wave * 16 + mrow;          // col within block's 128 cols

#ifdef GCN_HAVE_TDM
  const unsigned ldsA = (unsigned)(uintptr_t)(&As[0]);   // low 32b = LDS offset
  const unsigned ldsB = (unsigned)(uintptr_t)(&Bs[0]);
  const unsigned remRows = (unsigned)(M - mBase);        // OOB rows read as 0
#endif

  v8f acc = {0.f, 0.f, 0.f, 0.f, 0.f, 0.f, 0.f, 0.f};

  for (int kk = 0; kk < K; kk += KC) {
#ifdef GCN_HAVE_TDM
    if (tid < 32) {   // TDM ignores EXEC: issue from exactly one wave
      gcn_tdm_load_2d(H + (long long)mBase * K + kk, ldsA, KC, 16,
                      remRows < 16u ? remRows : 16u, (unsigned long long)K);
      gcn_tdm_load_2d(W + (long long)nBase * K + kk, ldsB, KC, 128, 128,
                      (unsigned long long)K);
      __builtin_amdgcn_s_wait_tensorcnt(0);
    }
#else
    // Fallback: cooperative staging into the same padded layout.
#pragma unroll
    for (int it = 0; it < (16 * KC) / 256; ++it) {
      const int idx = it * 256 + tid;
      const int r = idx >> 5, k = idx & 31;
      const int row = mBase + r;
      As[r * RS + k] = (row < M) ? H[(long long)row * K + kk + k] : 0.0f;
    }
#pragma unroll
    for (int it = 0; it < (KC * 128) / 256; ++it) {
      const int idx = it * 256 + tid;
      const int k = idx & 31, n = idx >> 5;
      Bs[n * RS + k] = W[(long long)(nBase + n) * K + kk + k];
    }
#endif
    __syncthreads();

    // f32 WMMA fragments (ISA 32-bit 16x4 layout):
    //   A: lanes 0-15 hold K=k0,k0+1 (M=lane); lanes 16-31 hold K=k0+2,k0+3.
    //   B mirrors with N=lane&15. Consecutive-K pairs -> single ds_load_b64.
#pragma unroll
    for (int k0 = 0; k0 < KC; k0 += 4) {
      v2f a, b;
      a[0] = As[mrow * RS + k0 + 2 * lh + 0];
      a[1] = As[mrow * RS + k0 + 2 * lh + 1];
      b[0] = Bs[ncol * RS + k0 + 2 * lh + 0];
      b[1] = Bs[ncol * RS + k0 + 2 * lh + 1];
      acc = __builtin_amdgcn_wmma_f32_16x16x4_f32(
          false, a, false, b, (short)0, acc, false, false);
    }
    __syncthreads();
  }

  // C/D layout: VGPR r -> M = r + 8*half, N = lane&15
  const int col = nBase + ncol;
#pragma unroll
  for (int r = 0; r < 8; ++r) {
    const int row = mBase + r + 8 * lh;
    if (row < M) {
      float val = acc[r];
      if (RELU) val = val > 0.0f ? val : 0.0f;
      O[(long long)row * NTOT + col] = val;
    }
  }
}

// ---------------------------------------------------------------------------
// Orchestration
// ---------------------------------------------------------------------------
extern "C" void kernel_launch(void* const* d_in, const int* in_sizes, int n_in,
                              void* d_out, int out_size, void* d_ws, size_t ws_size,
                              hipStream_t stream)
{
  const float* x  = (const float*)d_in[0];   // [M,128]
  const int*   er = (const int*)d_in[1];     // [nnz]
  const int*   ec = (const int*)d_in[2];     // [nnz]
  const float* ev = (const float*)d_in[3];   // [nnz]
  const float* W1 = (const float*)d_in[4];   // [256,128]
  const float* W2 = (const float*)d_in[5];   // [128,256]
  float* out = (float*)d_out;                // [M,128]

  const int M   = in_sizes[0] / 128;
  const int nnz = in_sizes[1];

  float* bufA = (float*)d_ws;                // M x 256 f32
  float* bufB = bufA + (long long)M * 256;   // M x 256 f32

  const long long n128 = (long long)M * 128;
  const long long n256 = (long long)M * 256;

  dim3 blk(256);
  dim3 sgrid((nnz + 7) / 8);
  auto zg4 = [](long long n) { return dim3((unsigned)((n / 4 + 255) / 256)); };

  // ---- layer 1: propagate x twice (d=128) ----
  gcn_zero_f32v4<<<zg4(n128), blk, 0, stream>>>((float4*)bufA, n128 / 4);
  gcn_spmm_scatter<128><<<sgrid, blk, 0, stream>>>(er, ec, ev, x, bufA, nnz);
  gcn_zero_f32v4<<<zg4(n128), blk, 0, stream>>>((float4*)bufB, n128 / 4);
  gcn_spmm_scatter<128><<<sgrid, blk, 0, stream>>>(er, ec, ev, bufA, bufB, nnz);

  // ---- relu(bufB @ W1^T) -> bufA [M,256] ----
  gcn_gemm_wmma<128, 256, true>
      <<<dim3((M + 15) / 16, 2), blk, 0, stream>>>(bufB, W1, bufA, M);

  // ---- layer 2: propagate twice (d=256) ----
  gcn_zero_f32v4<<<zg4(n256), blk, 0, stream>>>((float4*)bufB, n256 / 4);
  gcn_spmm_scatter<256><<<sgrid, blk, 0, stream>>>(er, ec, ev, bufA, bufB, nnz);
  gcn_zero_f32v4<<<zg4(n256), blk, 0, stream>>>((float4*)bufA, n256 / 4);
  gcn_spmm_scatter<256><<<sgrid, blk, 0, stream>>>(er, ec, ev, bufB, bufA, nnz);

  // ---- bufA @ W2^T -> out [M,128] ----
  gcn_gemm_wmma<256, 128, false>
      <<<dim3((M + 15) / 16, 1), blk, 0, stream>>>(bufA, W2, out, M);

  (void)n_in; (void)out_size; (void)ws_size;
}